// CCVGAE_12635793785673
// MI455X (gfx1250) — compile-verified
//
#include <hip/hip_runtime.h>
#include <hip/hip_bf16.h>
#include <math.h>

// ---------------------------------------------------------------------------
// CDNA5 (gfx1250) GAT-VGAE pipeline.
// GEMMs: V_WMMA_F32_16X16X4_F32 (full fp32 on the matrix pipes), A-tile staged
// via GLOBAL_LOAD_ASYNC_TO_LDS_B128 (double buffered, ASYNCcnt-tracked),
// B-tile pair-interleaved in LDS so b-fragments are single ds_load_b64s.
//   A 16x4 f32  : lane L holds row M=L%16, K = 2*(L/16)+{0,1} in 2 VGPRs
//   B 4x16 f32  : lane L holds col N=L%16, K = 2*(L/16)+{0,1} in 2 VGPRs
//   C/D 16x16   : VGPR r, lanes 0-15 -> M=r, N=lane ; lanes 16-31 -> M=r+8
// ---------------------------------------------------------------------------

typedef __attribute__((ext_vector_type(2))) float v2f;
typedef __attribute__((ext_vector_type(8))) float v8f;

#define BM 128            // rows per block (8 waves x 16)
#define BN 64             // cols per block (4 WMMA tiles per wave)
#define KB 16             // K chunk staged in LDS
#define AS_STRIDE 20      // floats; row base 80B (16B aligned for async b128)
#define BS_STRIDE 160     // floats; interleaved pair-rows (KB/2 rows)
#define AS_SIZE (BM * AS_STRIDE)          // 2560 floats
#define BS_SIZE ((KB / 2) * BS_STRIDE)    // 1280 floats

__device__ __forceinline__ unsigned lds32(const void* p) {
    // AMDGPU generic pointers to LDS carry the LDS byte address in addr[31:0].
    return (unsigned)(unsigned long long)p;
}
__device__ __forceinline__ void async_load_b128(unsigned lds_byte, unsigned gbyte_off,
                                                unsigned long long base) {
    asm volatile("global_load_async_to_lds_b128 %0, %1, %2"
                 :: "v"(lds_byte), "v"(gbyte_off), "s"(base) : "memory");
}
__device__ __forceinline__ void wait_async0() {
    asm volatile("s_wait_asynccnt 0x0" ::: "memory");
}

// C[M,N] = act(A[M,K] @ B[K,N] + bias)   (bias may be null; act: 0=none 1=relu)
// Requires K % 16 == 0 (true here: 2000, 256, 64, 16).
__global__ __launch_bounds__(256) void wmma_gemm_f32(
    const float* __restrict__ A, const float* __restrict__ B,
    const float* __restrict__ bias, float* __restrict__ C,
    int M, int N, int K, int act)
{
    __shared__ float As[2][AS_SIZE];
    __shared__ float Bs[2][BS_SIZE];

    const int tid  = threadIdx.x;
    const int wave = tid >> 5;            // 0..7
    const int lane = tid & 31;
    const int l16  = lane & 15;
    const int lhi  = lane >> 4;           // 0 or 1
    const int row0 = blockIdx.y * BM;
    const int col0 = blockIdx.x * BN;

    const unsigned long long Abase = (unsigned long long)A;

    // --- per-thread staging geometry ---
    const int ar   = tid >> 1;                  // A row within tile (0..127)
    const int ac   = (tid & 1) * 8;             // A col chunk (floats)
    const int agr  = row0 + ar;
    const int br   = tid >> 4;                  // B row within chunk (0..15)
    const int bc   = (tid & 15) * 2;            // B col (0..30)
    const int bidx = (br >> 1) * BS_STRIDE + (br & 1);

    v8f acc[4] = {};

    // issue A async loads for chunk k0 into buffer buf
    auto issue_a = [&](int k0, int buf) {
        if (agr < M) {
            unsigned lds = lds32(&As[buf][ar * AS_STRIDE + ac]);
            unsigned gof = (unsigned)(((size_t)agr * K + k0 + ac) * sizeof(float));
            async_load_b128(lds,      gof,      Abase);
            async_load_b128(lds + 16, gof + 16, Abase);
        }
    };
    // load B chunk k0 into registers (guarded, zero-filled)
    auto load_b = [&](int k0, float2& v0, float2& v1) {
        v0 = make_float2(0.f, 0.f);
        v1 = make_float2(0.f, 0.f);
        const float* p = B + (size_t)(k0 + br) * N + col0 + bc;
        if (col0 + bc + 1      < N) v0 = *(const float2*)p;
        if (col0 + bc + 32 + 1 < N) v1 = *(const float2*)(p + 32);
    };
    // store staged B registers pair-interleaved into buffer buf
    auto store_b = [&](int buf, float2 v0, float2 v1) {
        float* q = &Bs[buf][bidx];
        q[2 * bc]            = v0.x;
        q[2 * (bc + 1)]      = v0.y;
        q[2 * (bc + 32)]     = v1.x;
        q[2 * (bc + 33)]     = v1.y;
    };

    // --- prologue: stage chunk 0 ---
    {
        float2 v0, v1;
        issue_a(0, 0);
        load_b(0, v0, v1);
        store_b(0, v0, v1);
        wait_async0();
        __syncthreads();
    }

    int cur = 0;
    for (int k0 = 0; k0 < K; k0 += KB) {
        const int nxt = cur ^ 1;
        const bool more = (k0 + KB) < K;
        float2 nb0, nb1;
        if (more) {                       // overlap next-chunk fetch with compute
            issue_a(k0 + KB, nxt);
            load_b(k0 + KB, nb0, nb1);
        }

        // ---- compute on current buffers: 4 K-steps x 4 tiles = 16 WMMAs ----
        const float* arow = &As[cur][(wave * 16 + l16) * AS_STRIDE];
        #pragma unroll
        for (int kk = 0; kk < KB; kk += 4) {
            const int klo = kk + 2 * lhi;
            v2f a = *(const v2f*)&arow[klo];
            const float* brow = &Bs[cur][((kk >> 1) + lhi) * BS_STRIDE + 2 * l16];
            #pragma unroll
            for (int j = 0; j < 4; ++j) {
                v2f b = *(const v2f*)&brow[32 * j];
                acc[j] = __builtin_amdgcn_wmma_f32_16x16x4_f32(
                    false, a, false, b, (short)0, acc[j], false, false);
            }
        }

        if (more) store_b(nxt, nb0, nb1);
        wait_async0();
        __syncthreads();
        cur = nxt;
    }

    // ---- epilogue: bias + activation, guarded stores ----
    const int crow = row0 + wave * 16 + lhi * 8;
    #pragma unroll
    for (int j = 0; j < 4; ++j) {
        const int ccol = col0 + 16 * j + l16;
        float bv = 0.f;
        if (bias && ccol < N) bv = bias[ccol];
        if (ccol < N) {
            #pragma unroll
            for (int r = 0; r < 8; ++r) {
                int grow = crow + r;
                if (grow < M) {
                    float v = acc[j][r] + bv;
                    if (act == 1) v = fmaxf(v, 0.f);
                    C[(size_t)grow * N + ccol] = v;
                }
            }
        }
    }
}

// ---------------------------------------------------------------------------
// GAT attention helpers
// ---------------------------------------------------------------------------

__global__ void node_dual_dot(const float* __restrict__ h,
                              const float* __restrict__ asrc,
                              const float* __restrict__ adst,
                              float* __restrict__ hs, float* __restrict__ hd,
                              int N, int K)
{
    int node = blockIdx.x * (blockDim.x >> 5) + (threadIdx.x >> 5);
    int lane = threadIdx.x & 31;
    if (node >= N) return;
    const float* row = h + (size_t)node * K;
    float s0 = 0.f, s1 = 0.f;
    for (int c = lane; c < K; c += 32) {
        float v = row[c];
        s0 += v * asrc[c];
        s1 += v * adst[c];
    }
    #pragma unroll
    for (int off = 16; off > 0; off >>= 1) {
        s0 += __shfl_down(s0, off, 32);
        s1 += __shfl_down(s1, off, 32);
    }
    if (lane == 0) { hs[node] = s0; hd[node] = s1; }
}

__global__ void init_attn(int* __restrict__ m_int, float* __restrict__ ssum, int N)
{
    int i = blockIdx.x * blockDim.x + threadIdx.x;
    if (i < N) { m_int[i] = (int)0x80000000; ssum[i] = 0.f; }
}

__device__ __forceinline__ int flip_f32(float f) {
    int i = __float_as_int(f);
    return i >= 0 ? i : (i ^ 0x7fffffff);
}
__device__ __forceinline__ float unflip_f32(int i) {
    return __int_as_float(i >= 0 ? i : (i ^ 0x7fffffff));
}

__global__ void edge_score(const long long* __restrict__ ei,
                           const float* __restrict__ hs, const float* __restrict__ hd,
                           float* __restrict__ e, int* __restrict__ m_int, int E)
{
    int i = blockIdx.x * blockDim.x + threadIdx.x;
    if (i >= E) return;
    int s = (int)ei[i];
    int d = (int)ei[(size_t)E + i];
    float v = hs[s] + hd[d];
    v = v > 0.f ? v : 0.2f * v;
    e[i] = v;
    atomicMax(&m_int[d], flip_f32(v));
}

__global__ void edge_exp(const long long* __restrict__ ei,
                         const float* __restrict__ e, const int* __restrict__ m_int,
                         float* __restrict__ ee, float* __restrict__ ssum, int E)
{
    int i = blockIdx.x * blockDim.x + threadIdx.x;
    if (i >= E) return;
    int d = (int)ei[(size_t)E + i];
    float x = expf(e[i] - unflip_f32(m_int[d]));
    ee[i] = x;
    atomicAdd(&ssum[d], x);
}

__global__ void row_bias_init(float* __restrict__ out, const float* __restrict__ bias,
                              int total, int H)
{
    int i = blockIdx.x * blockDim.x + threadIdx.x;
    if (i < total) out[i] = bias[i % H];
}

__global__ void edge_scatter(const long long* __restrict__ ei,
                             const float* __restrict__ h,
                             const float* __restrict__ ee, const float* __restrict__ ssum,
                             float* __restrict__ out, int E, int H)
{
    int eidx = blockIdx.x * (blockDim.x >> 5) + (threadIdx.x >> 5);
    int lane = threadIdx.x & 31;
    if (eidx >= E) return;
    int s = (int)ei[eidx];
    int d = (int)ei[(size_t)E + eidx];
    float alpha = ee[eidx] / (ssum[d] + 1e-16f);
    const float* hrow = h + (size_t)s * H;
    float* orow = out + (size_t)d * H;
    for (int c = lane; c < H; c += 32)
        atomicAdd(&orow[c], hrow[c] * alpha);
}

// ---------------------------------------------------------------------------
// BatchNorm (column stats over N), rsample, misc
// ---------------------------------------------------------------------------

__global__ void zero_f32(float* __restrict__ p, int n)
{
    int i = blockIdx.x * blockDim.x + threadIdx.x;
    if (i < n) p[i] = 0.f;
}

__global__ void bn_stats(const float* __restrict__ x,
                         float* __restrict__ colsum, float* __restrict__ colsq,
                         int N, int H)
{
    int col = threadIdx.x;
    float s = 0.f, q = 0.f;
    for (int r = blockIdx.x; r < N; r += gridDim.x) {
        float v = x[(size_t)r * H + col];
        s += v;
        q += v * v;
    }
    atomicAdd(&colsum[col], s);
    atomicAdd(&colsq[col],  q);
}

__global__ void bn_apply(const float* __restrict__ x,
                         const float* __restrict__ colsum, const float* __restrict__ colsq,
                         const float* __restrict__ g, const float* __restrict__ be,
                         const float* __restrict__ residual,
                         float* __restrict__ out, int N, int H)
{
    int i = blockIdx.x * blockDim.x + threadIdx.x;
    if (i >= N * H) return;
    int col = i % H;
    float invN = 1.f / (float)N;
    float mu  = colsum[col] * invN;
    float var = colsq[col] * invN - mu * mu;
    float v = (x[i] - mu) * rsqrtf(var + 1e-5f) * g[col] + be[col];
    v = fmaxf(v, 0.f);
    if (residual) v += residual[i];
    out[i] = v;
}

__global__ void rsample_k(const float* __restrict__ qm, const float* __restrict__ qs,
                          const float* __restrict__ eps, float* __restrict__ qz, int total)
{
    int i = blockIdx.x * blockDim.x + threadIdx.x;
    if (i >= total) return;
    float x = qs[i];
    float sp = log1pf(expf(-fabsf(x))) + fmaxf(x, 0.f);
    qz[i] = qm[i] + (sp + 1e-6f) * eps[i];
}

// ---------------------------------------------------------------------------
// Host orchestration
// ---------------------------------------------------------------------------

static inline void launch_gemm(const float* A, const float* B, const float* bias,
                               float* C, int M, int N, int K, int act, hipStream_t s)
{
    dim3 grid((N + BN - 1) / BN, (M + BM - 1) / BM);
    wmma_gemm_f32<<<grid, 256, 0, s>>>(A, B, bias, C, M, N, K, act);
}

static void run_gat(const float* hin, int Kin, const float* W,
                    const float* asrc, const float* adst, const float* b, int Hout,
                    const long long* ei, int E, int N,
                    float* hraw, float* agg,
                    float* hs, float* hd, int* m_int, float* ssum,
                    float* ebuf, float* eebuf, hipStream_t s)
{
    launch_gemm(hin, W, nullptr, hraw, N, Hout, Kin, 0, s);
    node_dual_dot<<<(N + 7) / 8, 256, 0, s>>>(hraw, asrc, adst, hs, hd, N, Hout);
    init_attn<<<(N + 255) / 256, 256, 0, s>>>(m_int, ssum, N);
    edge_score<<<(E + 255) / 256, 256, 0, s>>>(ei, hs, hd, ebuf, m_int, E);
    edge_exp<<<(E + 255) / 256, 256, 0, s>>>(ei, ebuf, m_int, eebuf, ssum, E);
    row_bias_init<<<(N * Hout + 255) / 256, 256, 0, s>>>(agg, b, N * Hout, Hout);
    edge_scatter<<<(E + 7) / 8, 256, 0, s>>>(ei, hraw, eebuf, ssum, agg, E, Hout);
}

extern "C" void kernel_launch(void* const* d_in, const int* in_sizes, int n_in,
                              void* d_out, int out_size, void* d_ws, size_t ws_size,
                              hipStream_t stream)
{
    (void)n_in; (void)out_size; (void)ws_size;

    const float*     x      = (const float*)d_in[0];
    const long long* ei     = (const long long*)d_in[1];
    const float*     eps    = (const float*)d_in[2];
    const float *W0 = (const float*)d_in[3],  *as0 = (const float*)d_in[4],
                *ad0 = (const float*)d_in[5], *b0 = (const float*)d_in[6],
                *g0 = (const float*)d_in[7],  *be0 = (const float*)d_in[8];
    const float *W1 = (const float*)d_in[9],  *as1 = (const float*)d_in[10],
                *ad1 = (const float*)d_in[11],*b1 = (const float*)d_in[12],
                *g1 = (const float*)d_in[13], *be1 = (const float*)d_in[14];
    const float *Wm = (const float*)d_in[15], *asm_ = (const float*)d_in[16],
                *adm = (const float*)d_in[17],*bm = (const float*)d_in[18];
    const float *Wv = (const float*)d_in[19], *asv = (const float*)d_in[20],
                *adv = (const float*)d_in[21],*bv = (const float*)d_in[22];
    const float *leW = (const float*)d_in[23], *leb = (const float*)d_in[24];
    const float *ldW = (const float*)d_in[25], *ldb = (const float*)d_in[26];
    const float *dW0 = (const float*)d_in[27], *db0 = (const float*)d_in[28];
    const float *dW1 = (const float*)d_in[29], *db1 = (const float*)d_in[30];
    const float *dW2 = (const float*)d_in[31], *db2 = (const float*)d_in[32];

    const int H   = in_sizes[4];              // 256
    const int L   = in_sizes[16];             // 64
    const int I_  = in_sizes[24];             // 16
    const int DEC = in_sizes[28];             // 256
    const int D   = in_sizes[32];             // 2000
    const int N   = in_sizes[0] / D;          // 20000
    const int E   = in_sizes[1] / 2;          // 320000

    float* ws0   = (float*)d_ws;              // N*H  (raw GEMM outs / d0)
    float* ws1   = ws0 + (size_t)N * H;       // N*H  (aggregation / h2)
    float* ws2   = ws1 + (size_t)N * H;       // N*H  (h0 residual / d1)
    float* hs    = ws2 + (size_t)N * H;       // N
    float* hd    = hs + N;                    // N
    int*   m_int = (int*)(hd + N);            // N
    float* ssum  = (float*)(m_int + N);       // N
    float* ebuf  = ssum + N;                  // E
    float* eebuf = ebuf + E;                  // E
    float* colsum = eebuf + E;                // H
    float* colsq  = colsum + H;               // H

    float* out   = (float*)d_out;             // q_z,q_m,q_s,l_e,l_d,x_recon
    float* o_qz  = out;
    float* o_qm  = o_qz + (size_t)N * L;
    float* o_qs  = o_qm + (size_t)N * L;
    float* o_le  = o_qs + (size_t)N * L;
    float* o_ld  = o_le + (size_t)N * I_;
    float* o_xr  = o_ld + (size_t)N * L;

    // ===== GAT layer 0: x[D] -> h0[H] =====
    run_gat(x, D, W0, as0, ad0, b0, H, ei, E, N,
            ws0, ws1, hs, hd, m_int, ssum, ebuf, eebuf, stream);
    zero_f32<<<(2 * H + 255) / 256, 256, 0, stream>>>(colsum, 2 * H);
    bn_stats<<<256, H, 0, stream>>>(ws1, colsum, colsq, N, H);
    bn_apply<<<(N * H + 255) / 256, 256, 0, stream>>>(ws1, colsum, colsq, g0, be0,
                                                      nullptr, ws2, N, H);   // h0 -> ws2

    // ===== GAT layer 1: h0 -> h2 (+residual) =====
    run_gat(ws2, H, W1, as1, ad1, b1, H, ei, E, N,
            ws0, ws1, hs, hd, m_int, ssum, ebuf, eebuf, stream);
    zero_f32<<<(2 * H + 255) / 256, 256, 0, stream>>>(colsum, 2 * H);
    bn_stats<<<256, H, 0, stream>>>(ws1, colsum, colsq, N, H);
    bn_apply<<<(N * H + 255) / 256, 256, 0, stream>>>(ws1, colsum, colsq, g1, be1,
                                                      ws2, ws1, N, H);       // h2 -> ws1

    // ===== heads: q_m, q_s (GATConv H->L) =====
    run_gat(ws1, H, Wm, asm_, adm, bm, L, ei, E, N,
            ws0, o_qm, hs, hd, m_int, ssum, ebuf, eebuf, stream);
    run_gat(ws1, H, Wv, asv, adv, bv, L, ei, E, N,
            ws0, o_qs, hs, hd, m_int, ssum, ebuf, eebuf, stream);

    // ===== rsample =====
    rsample_k<<<(N * L + 255) / 256, 256, 0, stream>>>(o_qm, o_qs, eps, o_qz, N * L);

    // ===== coupling =====
    launch_gemm(o_qz, leW, leb, o_le, N, I_, L, 0, stream);
    launch_gemm(o_le, ldW, ldb, o_ld, N, L, I_, 0, stream);

    // ===== decoder =====
    launch_gemm(o_ld, dW0, db0, ws0, N, DEC, L, 1, stream);
    launch_gemm(ws0, dW1, db1, ws2, N, DEC, DEC, 1, stream);
    launch_gemm(ws2, dW2, db2, o_xr, N, D, DEC, 0, stream);
}